// multiHeadAttention_12103217840495
// MI455X (gfx1250) — compile-verified
//
#include <hip/hip_runtime.h>

// ---------------------------------------------------------------------------
// Multi-head attention with sparsemax, MI455X (gfx1250, wave32).
// - All GEMMs through one tiled WMMA-f32 kernel (v_wmma_f32_16x16x4_f32).
// - Global->LDS staging uses CDNA5 async copies (GLOBAL_LOAD_ASYNC_TO_LDS_B32,
//   ASYNCcnt) with LDS double buffering: chunk c+1 streams in while chunk c
//   computes; s_wait_asynccnt 0 + workgroup barrier close each iteration.
// - attn (512MB, >> 192MB L2) is written with non-temporal stores.
// ---------------------------------------------------------------------------

typedef __attribute__((ext_vector_type(2))) float v2f;
typedef __attribute__((ext_vector_type(8))) float v8f;

#define BM 64
#define BN 64
#define BK 32
#define LDA_S 33   // padded LDS strides to dodge bank conflicts
#define LDB_S 65
#define ASZ (BM * LDA_S)
#define BSZ (BK * LDB_S)

// Per-lane async copy: 4 bytes from global gbase + goffBytes to the LDS byte address ldsByte
// GVS addressing: 64-bit SGPR base plus 32-bit VGPR byte offset. Tracked by ASYNCcnt.
__device__ __forceinline__ void async_copy_b32(unsigned ldsByte,
                                               const float* gbase,
                                               unsigned goffBytes)
{
    asm volatile("global_load_async_to_lds_b32 %0, %1, %2"
                 :
                 : "v"(ldsByte), "v"(goffBytes), "s"(gbase)
                 : "memory");
}

__device__ __forceinline__ void wait_async_all()
{
    asm volatile("s_wait_asynccnt 0x0" ::: "memory");
}

// Stage one BMxBK A-tile and BKxBN B-tile into the given LDS buffers with
// async copies (issued only; caller waits).
__device__ __forceinline__ void stage_tiles_async(
    unsigned asByte, unsigned bsByte,
    const float* __restrict__ A, int lda,
    const float* __restrict__ B, int ldb, int transB,
    int mBase, int nBase, int k0, int tid)
{
    #pragma unroll
    for (int i = 0; i < (BM * BK) / 256; ++i) {
        int lid = tid + i * 256;
        int mm = lid >> 5, kk = lid & 31;                     // coalesced along k
        unsigned l = asByte + (unsigned)(mm * LDA_S + kk) * 4u;
        unsigned g = (unsigned)((mBase + mm) * lda + (k0 + kk)) * 4u;
        async_copy_b32(l, A, g);
    }
    if (transB) {
        #pragma unroll
        for (int i = 0; i < (BK * BN) / 256; ++i) {
            int lid = tid + i * 256;
            int nn = lid >> 5, kk = lid & 31;                 // coalesced along k
            unsigned l = bsByte + (unsigned)(kk * LDB_S + nn) * 4u;
            unsigned g = (unsigned)((nBase + nn) * ldb + (k0 + kk)) * 4u;
            async_copy_b32(l, B, g);
        }
    } else {
        #pragma unroll
        for (int i = 0; i < (BK * BN) / 256; ++i) {
            int lid = tid + i * 256;
            int kk = lid >> 6, nn = lid & 63;                 // coalesced along n
            unsigned l = bsByte + (unsigned)(kk * LDB_S + nn) * 4u;
            unsigned g = (unsigned)((k0 + kk) * ldb + (nBase + nn)) * 4u;
            async_copy_b32(l, B, g);
        }
    }
}

// Load one A-fragment (16x4) and two B-fragments (4x16) from LDS.
// ISA 7.12.2 f32 layouts: lanes 0-15 hold K=0,1; lanes 16-31 hold K=2,3.
__device__ __forceinline__ void load_frags(const float* __restrict__ Asb,
                                           const float* __restrict__ Bsb,
                                           int m0w, int n0w, int fr, int fk,
                                           int kk, v2f& a, v2f& b0, v2f& b1)
{
    const float* ap = Asb + (m0w + fr) * LDA_S + kk + fk;
    a.x = ap[0];
    a.y = ap[1];
    const float* bp = Bsb + (kk + fk) * LDB_S + n0w + fr;
    b0.x = bp[0];
    b0.y = bp[LDB_S];
    b1.x = bp[16];
    b1.y = bp[LDB_S + 16];
}

// C[M,N] = scale * (A[M,K] x B) + bias[N]
//   transB=1 : B is W[N][K] row-major (A·Wᵀ);  transB=0 : B is B[K][N] row-major
// blockIdx.z selects a head; a/b/cHS are per-head element strides.
// All dims are multiples of the tile sizes. ntStore=1 -> non-temporal C stores.
__global__ __launch_bounds__(256) void wmma_gemm_f32(
    const float* __restrict__ A, int lda, long long aHS,
    const float* __restrict__ B, int ldb, long long bHS, int transB,
    const float* __restrict__ bias,
    float* __restrict__ C, int ldc, long long cHS,
    int Kdim, float scale, int ntStore)
{
    __shared__ float As[2 * ASZ];
    __shared__ float Bs[2 * BSZ];

    const int tid  = threadIdx.x;
    const int lane = tid & 31;
    const int wave = tid >> 5;
    const int m0w  = (wave & 3) << 4;   // 4 waves down M
    const int n0w  = (wave >> 2) << 5;  // 2 waves across N, 32 cols each
    const int mBase = blockIdx.y * BM;
    const int nBase = blockIdx.x * BN;

    A += (long long)blockIdx.z * aHS;
    B += (long long)blockIdx.z * bHS;
    C += (long long)blockIdx.z * cHS;

    const unsigned asByte = (unsigned)(unsigned long long)(const void*)As;
    const unsigned bsByte = (unsigned)(unsigned long long)(const void*)Bs;

    v8f acc0 = {};
    v8f acc1 = {};

    const int fr = lane & 15;           // A row / B col / C col within 16x16 tile
    const int fk = (lane >> 4) << 1;    // upper half-wave holds K=2,3

    const int nCh = Kdim / BK;

    // Prologue: stream chunk 0 into buffer 0.
    stage_tiles_async(asByte, bsByte, A, lda, B, ldb, transB, mBase, nBase, 0, tid);
    wait_async_all();
    __syncthreads();

    for (int c = 0; c < nCh; ++c) {
        const int cur = c & 1;
        const float* Asb = As + cur * ASZ;
        const float* Bsb = Bs + cur * BSZ;

        // Kick off the next chunk's async copies into the other buffer.
        if (c + 1 < nCh) {
            stage_tiles_async(asByte + (unsigned)((cur ^ 1) * ASZ) * 4u,
                              bsByte + (unsigned)((cur ^ 1) * BSZ) * 4u,
                              A, lda, B, ldb, transB,
                              mBase, nBase, (c + 1) * BK, tid);
        }

        // Compute on the current buffer; 2-stage pipelined fragment loads.
        v2f a_c, b0_c, b1_c;
        load_frags(Asb, Bsb, m0w, n0w, fr, fk, 0, a_c, b0_c, b1_c);
        #pragma unroll
        for (int kk = 0; kk < BK; kk += 4) {
            v2f a_n, b0_n, b1_n;
            if (kk + 4 < BK)
                load_frags(Asb, Bsb, m0w, n0w, fr, fk, kk + 4, a_n, b0_n, b1_n);
            acc0 = __builtin_amdgcn_wmma_f32_16x16x4_f32(false, a_c, false, b0_c,
                                                         (short)0, acc0, false, false);
            acc1 = __builtin_amdgcn_wmma_f32_16x16x4_f32(false, a_c, false, b1_c,
                                                         (short)0, acc1, false, false);
            a_c = a_n; b0_c = b0_n; b1_c = b1_n;
        }

        // Own async copies done + everyone finished reading before buffer swap.
        wait_async_all();
        __syncthreads();
    }

    // C/D layout: VGPR r holds M=r (lanes 0-15) and M=r+8 (lanes 16-31), N=lane&15
    const int col0 = nBase + n0w + fr;
    const int row0 = mBase + m0w + ((lane >> 4) << 3);
    const float bv0 = bias ? bias[col0]      : 0.0f;
    const float bv1 = bias ? bias[col0 + 16] : 0.0f;
    if (ntStore) {
        #pragma unroll
        for (int r = 0; r < 8; ++r) {
            __builtin_nontemporal_store(acc0[r] * scale + bv0,
                                        &C[(long long)(row0 + r) * ldc + col0]);
            __builtin_nontemporal_store(acc1[r] * scale + bv1,
                                        &C[(long long)(row0 + r) * ldc + col0 + 16]);
        }
    } else {
        #pragma unroll
        for (int r = 0; r < 8; ++r) {
            C[(long long)(row0 + r) * ldc + col0]      = acc0[r] * scale + bv0;
            C[(long long)(row0 + r) * ldc + col0 + 16] = acc1[r] * scale + bv1;
        }
    }
}

// ---------------------------------------------------------------------------
// sparsemax over rows of 4096: one block (8 wave32 waves) per row.
// tau solves sum(max(z - tau, 0)) == 1; monotone in tau, bracketed by
// [zmax - 1, zmax), 40 bisection steps (2^-40 interval << fp32 ulp here).
// ---------------------------------------------------------------------------
#define NKROW 4096
__global__ __launch_bounds__(256) void sparsemax_kernel(float* __restrict__ attn)
{
    __shared__ float red[9];
    const int tid = threadIdx.x;
    float* z = attn + (long long)blockIdx.x * NKROW;

    float v[16];
    #pragma unroll
    for (int i = 0; i < 16; ++i) v[i] = z[tid + (i << 8)];

    // block max
    float m = v[0];
    #pragma unroll
    for (int i = 1; i < 16; ++i) m = fmaxf(m, v[i]);
    for (int o = 16; o > 0; o >>= 1) m = fmaxf(m, __shfl_down(m, o, 32));
    if ((tid & 31) == 0) red[tid >> 5] = m;
    __syncthreads();
    if (tid == 0) {
        float x = red[0];
        for (int i = 1; i < 8; ++i) x = fmaxf(x, red[i]);
        red[8] = x;
    }
    __syncthreads();
    const float zmax = red[8];

    float lo = zmax - 1.0f, hi = zmax;
    for (int it = 0; it < 40; ++it) {
        const float t = 0.5f * (lo + hi);
        float s = 0.0f;
        #pragma unroll
        for (int i = 0; i < 16; ++i) s += fmaxf(v[i] - t, 0.0f);
        for (int o = 16; o > 0; o >>= 1) s += __shfl_down(s, o, 32);
        __syncthreads();                 // protect red[] reuse across iterations
        if ((tid & 31) == 0) red[tid >> 5] = s;
        __syncthreads();
        if (tid == 0) {
            float x = 0.0f;
            for (int i = 0; i < 8; ++i) x += red[i];
            red[8] = x;
        }
        __syncthreads();
        const float total = red[8];      // uniform across block -> no divergence
        if (total >= 1.0f) lo = t; else hi = t;
    }
    const float tau = 0.5f * (lo + hi);
    #pragma unroll
    for (int i = 0; i < 16; ++i)
        __builtin_nontemporal_store(fmaxf(v[i] - tau, 0.0f), &z[tid + (i << 8)]);
}

// ---------------------------------------------------------------------------
// Host-side pipeline (graph-capture safe: launches only, all on `stream`).
// d_out = out[2048*1024] ++ attn[16*2048*4096]   (reference returns (out, attn))
// d_ws  = Q[2048*1024] ++ K[4096*1024] ++ V[4096*1024] ++ O[2048*1024] (48 MB)
// ---------------------------------------------------------------------------
extern "C" void kernel_launch(void* const* d_in, const int* in_sizes, int n_in,
                              void* d_out, int out_size, void* d_ws, size_t ws_size,
                              hipStream_t stream)
{
    (void)in_sizes; (void)n_in; (void)out_size; (void)ws_size;

    const float* enc = (const float*)d_in[0];
    const float* mem = (const float*)d_in[1];
    const float* Wq  = (const float*)d_in[2];
    const float* bq  = (const float*)d_in[3];
    const float* Wk  = (const float*)d_in[4];
    const float* bkb = (const float*)d_in[5];
    const float* Wv  = (const float*)d_in[6];
    const float* bvb = (const float*)d_in[7];
    const float* Wf  = (const float*)d_in[8];
    const float* bf  = (const float*)d_in[9];

    float* out  = (float*)d_out;
    float* attn = out + (size_t)2048 * 1024;

    float* Qb = (float*)d_ws;
    float* Kb = Qb + (size_t)2048 * 1024;
    float* Vb = Kb + (size_t)4096 * 1024;
    float* Ob = Vb + (size_t)4096 * 1024;

    const long long attnHS = (long long)2048 * 4096;
    dim3 blk(256);

    // Projections: X · Wᵀ + b  (K = 1024)
    wmma_gemm_f32<<<dim3(16, 32, 1), blk, 0, stream>>>(enc, 1024, 0, Wq, 1024, 0, 1, bq,  Qb, 1024, 0, 1024, 1.0f, 0);
    wmma_gemm_f32<<<dim3(16, 64, 1), blk, 0, stream>>>(mem, 1024, 0, Wk, 1024, 0, 1, bkb, Kb, 1024, 0, 1024, 1.0f, 0);
    wmma_gemm_f32<<<dim3(16, 64, 1), blk, 0, stream>>>(mem, 1024, 0, Wv, 1024, 0, 1, bvb, Vb, 1024, 0, 1024, 1.0f, 0);

    // Scores per head: S[h] = Q_h · K_hᵀ / sqrt(1024)  (K = 64, z = head, NT stores)
    wmma_gemm_f32<<<dim3(64, 32, 16), blk, 0, stream>>>(Qb, 1024, 64, Kb, 1024, 64, 1, nullptr,
                                                        attn, 4096, attnHS, 64, 0.03125f, 1);

    // sparsemax in place over each of the 16*2048 rows of length 4096
    sparsemax_kernel<<<dim3(16 * 2048), blk, 0, stream>>>(attn);

    // Context per head: O_h = attn_h · V_h   (K = 4096, B row-major)
    wmma_gemm_f32<<<dim3(1, 32, 16), blk, 0, stream>>>(attn, 4096, attnHS, Vb, 1024, 64, 0, nullptr,
                                                       Ob, 1024, 64, 4096, 1.0f, 0);

    // Final FC: out = O · Wfᵀ + bf
    wmma_gemm_f32<<<dim3(16, 32, 1), blk, 0, stream>>>(Ob, 1024, 0, Wf, 1024, 0, 1, bf,
                                                       out, 1024, 0, 1024, 1.0f, 0);
}